// cleanCEL_38783554682999
// MI455X (gfx1250) — compile-verified
//
#include <hip/hip_runtime.h>
#include <math.h>

#define TT 600
#define WAVES_PER_BLOCK 8
#define BIGIDX 0x3fffffff

typedef __attribute__((ext_vector_type(2))) float v2f;
typedef __attribute__((ext_vector_type(8))) float v8f;

__device__ __forceinline__ float wave_sum(float v) {
    for (int off = 16; off > 0; off >>= 1)
        v += __shfl_down(v, off, 32);
    return v; // valid at lane 0
}

// One wave per sample. y row is staged into LDS with gfx1250 async
// global->LDS transfers (ASYNCcnt), scanned from LDS for the segment
// boundaries, then CE is summed only over the active intervals of ypred.
__global__ void per_sample_kernel(const float* __restrict__ ypred,
                                  const int* __restrict__ y,
                                  float* __restrict__ partials, int N) {
    __shared__ __align__(16) int   ybuf[WAVES_PER_BLOCK][TT];   // 19200 B
    __shared__ float blkSum[WAVES_PER_BLOCK];
    const int lane = threadIdx.x & 31;
    const int wave = threadIdx.x >> 5;
    const int n = blockIdx.x * WAVES_PER_BLOCK + wave;

    float per_sample = 0.0f;
    if (n < N) {
        const int* yr = y + (size_t)n * TT;
        const float* xr = ypred + (size_t)n * (TT * 2);

        // ---- stage y row (600 ints = 150 x b128) into this wave's LDS slice
        {
            const unsigned lds_base = (unsigned)(size_t)(&ybuf[wave][0]);
            #pragma unroll
            for (int i = 0; i < 5; ++i) {
                int idx = lane + i * 32;           // 16-byte chunk index
                if (idx < TT / 4) {
                    unsigned lds_off = lds_base + (unsigned)idx * 16u;
                    unsigned goff = (unsigned)idx * 16u;
                    asm volatile(
                        "global_load_async_to_lds_b128 %0, %1, %2"
                        :
                        : "v"(lds_off), "v"(goff), "s"((const void*)yr)
                        : "memory");
                }
            }
            asm volatile("s_wait_asynccnt 0x0" ::: "memory");
        }

        // ---- phase 1: first up-transition (s) / down-transition (e) from LDS
        int up = BIGIDX, dn = BIGIDX;
        for (int t = lane + 1; t < TT; t += 32) {
            int c = ybuf[wave][t];
            int p = ybuf[wave][t - 1];
            if (c == 1 && p == 0 && t < up) up = t;
            if (c == 0 && p == 1 && t < dn) dn = t;
        }
        for (int off = 16; off > 0; off >>= 1) {
            up = min(up, __shfl_down(up, off, 32));
            dn = min(dn, __shfl_down(dn, off, 32));
        }
        up = __shfl(up, 0, 32);
        dn = __shfl(dn, 0, 32);
        const int s = (up < BIGIDX) ? up : 0;
        const int e = (dn < BIGIDX) ? dn : TT;

        // ---- interval bounds (exact reference arithmetic)
        const int sz1    = min(s + 30, e - 10);
        const int szend1 = max(s + 40, e - 30);
        const int len1   = szend1 - sz1;
        const int e1     = max(1, s - 30);
        const int rem    = len1 - e1;
        const int s2     = min(TT - 1, szend1 + 30);
        const int e2     = min(TT, s2 + rem);
        const bool branchA = (len1 < s);

        int lo[3], hi[3];
        lo[0] = sz1; hi[0] = szend1;
        lo[1] = 0;   hi[1] = branchA ? len1 : e1;
        lo[2] = branchA ? 0 : s2;
        hi[2] = branchA ? 0 : e2;

        // prefetch the center interval (gfx1250 global_prefetch)
        {
            int off = min(2 * max(0, sz1) + lane * 16, 2 * TT - 1);
            __builtin_prefetch(xr + off, 0, 0);
        }

        float val = 0.0f;
        #pragma unroll
        for (int k = 0; k < 3; ++k) {
            const int l = max(lo[k], 0);
            const int h = min(hi[k], TT);
            const float cnt = (float)max(h - l, 1);
            float acc = 0.0f;
            for (int t = l + lane; t < h; t += 32) {
                v2f x = *(const v2f*)(xr + 2 * t);
                // labels are a single contiguous segment [s, e)
                const bool yt = (t >= s) && (t < e);
                // ce = -log_softmax(x)[yt]  (C=2, stable softplus form)
                const float d = yt ? (x.x - x.y) : (x.y - x.x);
                acc += fmaxf(d, 0.0f) + log1pf(expf(-fabsf(d)));
            }
            val += acc / cnt;  // cnt is wave-uniform
        }
        per_sample = wave_sum(val);  // lane 0 holds the sample's loss
    }

    if (lane == 0) blkSum[wave] = per_sample;
    __syncthreads();
    if (threadIdx.x == 0) {
        float ssum = 0.0f;
        #pragma unroll
        for (int w = 0; w < WAVES_PER_BLOCK; ++w) ssum += blkSum[w];
        partials[blockIdx.x] = ssum;  // fixed order -> deterministic
    }
}

// Final reduction: per-wave 64-value reduce via v_wmma_f32_16x16x4_f32
// against a ones matrix (exact f32 row sums), then tiny LDS combine.
__global__ void final_reduce_kernel(const float* __restrict__ partials,
                                    int nPartials, float* __restrict__ out,
                                    float invN) {
    __shared__ float wsum[8];
    const int tid = threadIdx.x;   // blockDim.x == 256
    const int lane = tid & 31;
    const int wave = tid >> 5;

    float a0 = 0.0f, a1 = 0.0f;
    for (int i = tid; i < nPartials; i += 512)       a0 += partials[i];
    for (int i = tid + 256; i < nPartials; i += 512) a1 += partials[i];

    v2f A; A.x = a0; A.y = a1;     // this wave's 64 accumulators = A (16x4)
    v2f B; B.x = 1.0f; B.y = 1.0f; // ones (4x16)
    v8f C = {};
    C = __builtin_amdgcn_wmma_f32_16x16x4_f32(
        /*neg_a=*/false, A, /*neg_b=*/false, B,
        /*c_mod=*/(short)0, C, /*reuse_a=*/false, /*reuse_b=*/false);

    float t = C[0] + C[1] + C[2] + C[3] + C[4] + C[5] + C[6] + C[7];
    t += __shfl_xor(t, 16, 32);    // fold rows 0-7 with rows 8-15

    if (lane == 0) wsum[wave] = t;
    __syncthreads();
    if (tid == 0) {
        float g = 0.0f;
        #pragma unroll
        for (int w = 0; w < 8; ++w) g += wsum[w];
        out[0] = g * invN;
    }
}

extern "C" void kernel_launch(void* const* d_in, const int* in_sizes, int n_in,
                              void* d_out, int out_size, void* d_ws, size_t ws_size,
                              hipStream_t stream) {
    const float* ypred = (const float*)d_in[0];   // [N, 600, 2] f32
    const int*   y     = (const int*)d_in[1];     // [N, 600] int
    float* out = (float*)d_out;
    float* partials = (float*)d_ws;

    const int N = in_sizes[1] / TT;
    const int blocks = (N + WAVES_PER_BLOCK - 1) / WAVES_PER_BLOCK;

    per_sample_kernel<<<blocks, WAVES_PER_BLOCK * 32, 0, stream>>>(ypred, y, partials, N);
    final_reduce_kernel<<<1, 256, 0, stream>>>(partials, blocks, out, 1.0f / (float)N);
}